// Sub_MGU_27582279975063
// MI455X (gfx1250) — compile-verified
//
#include <hip/hip_runtime.h>

typedef __attribute__((ext_vector_type(2))) float v2f;
typedef __attribute__((ext_vector_type(4))) float v4f;
typedef __attribute__((ext_vector_type(8))) float v8f;

#define B_ 8
#define T_ 1024
#define S_ 512
#define H_ 16
#define WAVES_PER_BLOCK 4
#define LDS_STRIDE 17  // 16 + 1 pad: dodge 64-bank conflicts on column reads

__device__ __forceinline__ float fast_sigmoid(float x) {
    // 1 / (1 + 2^(-x*log2e))  -> v_exp_f32 + v_rcp_f32
    return __builtin_amdgcn_rcpf(1.0f + __builtin_amdgcn_exp2f(-1.4426950408889634f * x));
}
__device__ __forceinline__ float fast_tanh(float x) {
    // tanh(x) = 2/(1+2^(-2x*log2e)) - 1
    return 2.0f * __builtin_amdgcn_rcpf(1.0f + __builtin_amdgcn_exp2f(-2.8853900817779268f * x)) - 1.0f;
}

__global__ __launch_bounds__(WAVES_PER_BLOCK * 32)
void mgu_scan_kernel(const float* __restrict__ x,       // (B,T,S)
                     const float* __restrict__ W_hir,   // (S,H)
                     const float* __restrict__ W_hii,   // (S,H)
                     const float* __restrict__ W_hin,   // (S,H)
                     const float* __restrict__ W_hhr,   // (S,H,H)
                     const float* __restrict__ bias_ih, // (3,S,H)
                     const float* __restrict__ bias_hh, // (3,S,H)
                     float* __restrict__ out)           // (B,T,S*H)
{
    __shared__ float lds[WAVES_PER_BLOCK][16 * LDS_STRIDE];

    const int wave = threadIdx.x >> 5;
    const int lane = threadIdx.x & 31;
    const int s    = blockIdx.x * WAVES_PER_BLOCK + wave;  // grid sized exactly: s < S_
    const int half = lane >> 4;   // 0: lanes 0-15, 1: lanes 16-31
    const int l16  = lane & 15;   // column index (batch) in B/C/D layouts

    float* tile = lds[wave];

    // ---- A fragments: A(M=h_out, K=h_in) = W_hhr[s][M][K], 16x4 f32 layout:
    // lanes 0-15: M=l16, VGPR0 K=4kk+0, VGPR1 K=4kk+1 ; lanes 16-31: K=4kk+2, 4kk+3
    const float* Wr = W_hhr + (size_t)s * H_ * H_;
    v2f afrag[4];
#pragma unroll
    for (int kk = 0; kk < 4; ++kk) {
        const int k0 = kk * 4 + half * 2;
        afrag[kk].x = Wr[l16 * H_ + k0];
        afrag[kk].y = Wr[l16 * H_ + k0 + 1];
    }

    // ---- per-row constants; D-layout row for VGPR r is h = r + 8*half
    float wr[8], wi[8], wn[8], cbr[8], cbi[8], cbni[8], cbnh[8];
#pragma unroll
    for (int r = 0; r < 8; ++r) {
        const int h  = r + 8 * half;
        const int sh = s * H_ + h;
        wr[r]   = W_hir[sh];
        wi[r]   = W_hii[sh];
        wn[r]   = W_hin[sh];
        cbr[r]  = bias_ih[0 * S_ * H_ + sh] + bias_hh[0 * S_ * H_ + sh];
        cbi[r]  = bias_ih[1 * S_ * H_ + sh] + bias_hh[1 * S_ * H_ + sh];
        cbni[r] = bias_ih[2 * S_ * H_ + sh];
        cbnh[r] = bias_hh[2 * S_ * H_ + sh];
    }

    // ---- state: hD = h in D layout (row M = h-index, col N = batch), init 0
    v8f hD = {};
    v2f bfrag[4];
#pragma unroll
    for (int kk = 0; kk < 4; ++kk) { bfrag[kk].x = 0.0f; bfrag[kk].y = 0.0f; }

    const bool active  = (l16 < B_);
    const int  b       = l16 & 7;  // clamped batch for address math
    const float* xp    = x + ((size_t)b * T_) * S_ + s;
    float*       outp  = out + ((size_t)b * T_) * (S_ * H_) + s * H_ + 8 * half;

    float xv = active ? xp[0] : 0.0f;

    for (int t = 0; t < T_; ++t) {
        // software prefetch of next timestep's x
        float xnext = 0.0f;
        if (active && (t + 1 < T_)) xnext = xp[(size_t)(t + 1) * S_];

        // gh^T (16x16, M=h_out, N=batch) = W_hhr[s] x h^T : 4 chained K=4 f32 WMMAs
        v8f g = {};
        g = __builtin_amdgcn_wmma_f32_16x16x4_f32(false, afrag[0], false, bfrag[0], (short)0, g, false, false);
        g = __builtin_amdgcn_wmma_f32_16x16x4_f32(false, afrag[1], false, bfrag[1], (short)0, g, false, false);
        g = __builtin_amdgcn_wmma_f32_16x16x4_f32(false, afrag[2], false, bfrag[2], (short)0, g, false, false);
        g = __builtin_amdgcn_wmma_f32_16x16x4_f32(false, afrag[3], false, bfrag[3], (short)0, g, false, false);

        // gate math, elementwise on D layout
#pragma unroll
        for (int r = 0; r < 8; ++r) {
            const float gh = g[r];
            const float rg = fast_sigmoid(__builtin_fmaf(xv, wr[r], cbr[r]) + gh);
            const float ig = fast_sigmoid(__builtin_fmaf(xv, wi[r], cbi[r]) + gh);
            const float ng = fast_tanh(__builtin_fmaf(xv, wn[r], cbni[r]) +
                                       rg * (gh + cbnh[r]));
            hD[r] = ng + ig * (hD[r] - ng);
        }

        // write h_new to out: 8 contiguous floats per active lane -> 2x b128 stores
        if (active) {
            float* op = outp + (size_t)t * (S_ * H_);
            v4f o0 = { hD[0], hD[1], hD[2], hD[3] };
            v4f o1 = { hD[4], hD[5], hD[6], hD[7] };
            *(v4f*)(op)     = o0;
            *(v4f*)(op + 4) = o1;
        }

        // D-layout -> B-layout state transpose through LDS (same-wave DS is in-order)
#pragma unroll
        for (int r = 0; r < 8; ++r)
            tile[(r + 8 * half) * LDS_STRIDE + l16] = hD[r];
        asm volatile("" ::: "memory");  // keep ds_load after ds_store
#pragma unroll
        for (int kk = 0; kk < 4; ++kk) {
            const int k0 = kk * 4 + half * 2;
            bfrag[kk].x = tile[(k0)     * LDS_STRIDE + l16];
            bfrag[kk].y = tile[(k0 + 1) * LDS_STRIDE + l16];
        }

        xv = xnext;
    }
}

extern "C" void kernel_launch(void* const* d_in, const int* in_sizes, int n_in,
                              void* d_out, int out_size, void* d_ws, size_t ws_size,
                              hipStream_t stream) {
    (void)in_sizes; (void)n_in; (void)d_ws; (void)ws_size; (void)out_size;
    const float* x       = (const float*)d_in[0];
    const float* W_hir   = (const float*)d_in[1];
    const float* W_hii   = (const float*)d_in[2];
    const float* W_hin   = (const float*)d_in[3];
    const float* W_hhr   = (const float*)d_in[4];
    const float* bias_ih = (const float*)d_in[5];
    const float* bias_hh = (const float*)d_in[6];
    float* out = (float*)d_out;

    dim3 grid(S_ / WAVES_PER_BLOCK);          // 128 blocks
    dim3 block(WAVES_PER_BLOCK * 32);         // 4 waves/block, one wave per s
    mgu_scan_kernel<<<grid, block, 0, stream>>>(x, W_hir, W_hii, W_hin, W_hhr,
                                                bias_ih, bias_hh, out);
}